// MemorizingLayer_29746943492382
// MI455X (gfx1250) — compile-verified
//
#include <hip/hip_runtime.h>
#include <math.h>

typedef __attribute__((ext_vector_type(2))) float v2f;
typedef __attribute__((ext_vector_type(8))) float v8f;
typedef __attribute__((__vector_size__(4 * sizeof(int)))) int v4i_;

// D = A(16x4, f32) * B(4x16, f32) + C(16x16, f32), full fp32 WMMA on gfx1250
#define WMMA_F32X4(a, b, c) \
  __builtin_amdgcn_wmma_f32_16x16x4_f32(false, (a), false, (b), (short)0, (c), false, false)

#if __has_builtin(__builtin_amdgcn_global_load_async_to_lds_b128) && \
    __has_builtin(__builtin_amdgcn_s_wait_asynccnt)
#define USE_ASYNC_LDS 1
#else
#define USE_ASYNC_LDS 0
#endif

__device__ __forceinline__ void cp_b128(float* lds_dst, const float* gsrc) {
#if USE_ASYNC_LDS
  __builtin_amdgcn_global_load_async_to_lds_b128(
      (__attribute__((address_space(1))) v4i_*)gsrc,
      (__attribute__((address_space(3))) v4i_*)lds_dst, 0, 0);
#else
  *(float4*)lds_dst = *(const float4*)gsrc;
#endif
}

constexpr int Bz = 4, Sq = 1024, Dm = 1024, Hh = 16, DHd = 64, Mm = 4096, Kt = 32, Ff = 4096;

// ---------------------------------------------------------------------------
// WMMA GEMM v2: C[M,N] = A[M,K] @ B[K,N] (+bias, +relu)
// block = 256 threads (8 waves). Block tile 64x128, BK=32, double-buffered LDS.
// Wave (wr,wc) of 2x4 computes a 32x32 sub-tile (4 WMMA accumulators).
// Tiles staged via async global->LDS (ASYNCcnt) when available.
// ---------------------------------------------------------------------------
#define A_STRIDE 36   // 32 + 4 pad (floats), keeps 16B alignment
#define B_STRIDE 132  // 128 + 4 pad (floats)

__global__ __launch_bounds__(256) void gemm_wmma_f32(
    const float* __restrict__ A, const float* __restrict__ Bm,
    const float* __restrict__ bias, float* __restrict__ C,
    int Md, int Nd, int Kd, int do_relu)
{
  __shared__ float sa[2][64 * A_STRIDE];
  __shared__ float sb[2][32 * B_STRIDE];

  const int tid = threadIdx.x;
  const int w = tid >> 5, lane = tid & 31;
  const int lm = lane & 15, hf = lane >> 4;
  const int wr = w >> 2, wc = w & 3;              // 2 x 4 wave grid
  const int row0 = blockIdx.x * 64;
  const int colB = blockIdx.y * 128;

  // per-thread copy slots
  const int ra = tid >> 3, kq = tid & 7;          // A: rows ra, ra+32; 4-float col kq*4
  const int rb = tid >> 5, nq = tid & 31;         // B: rows rb+8j;     4-float col nq*4
  const int la0 = ra * A_STRIDE + kq * 4;
  const int la1 = (ra + 32) * A_STRIDE + kq * 4;

  const float* pA = A + (size_t)(row0 + ra) * Kd + kq * 4;
  const float* pB = Bm + (size_t)rb * Nd + colB + nq * 4;

  v8f acc00 = {}, acc01 = {}, acc10 = {}, acc11 = {};

  const int nt = Kd / 32;
  int cur = 0;

  // prologue: issue tile 0 into buffer 0
  {
    cp_b128(&sa[0][la0], pA);
    cp_b128(&sa[0][la1], pA + 32 * (size_t)Kd);
#pragma unroll
    for (int j = 0; j < 4; ++j)
      cp_b128(&sb[0][(rb + 8 * j) * B_STRIDE + nq * 4], pB + (size_t)(8 * j) * Nd);
    pA += 32;
    pB += (size_t)32 * Nd;
  }

  for (int it = 0; it < nt; ++it) {
    const bool has_next = (it + 1 < nt);
    if (has_next) {
      const int nxt = cur ^ 1;
      cp_b128(&sa[nxt][la0], pA);
      cp_b128(&sa[nxt][la1], pA + 32 * (size_t)Kd);
#pragma unroll
      for (int j = 0; j < 4; ++j)
        cp_b128(&sb[nxt][(rb + 8 * j) * B_STRIDE + nq * 4], pB + (size_t)(8 * j) * Nd);
      pA += 32;
      pB += (size_t)32 * Nd;
    }
#if USE_ASYNC_LDS
    if (has_next) __builtin_amdgcn_s_wait_asynccnt(6);  // current tile's 6 ops done
    else          __builtin_amdgcn_s_wait_asynccnt(0);
#endif
    __syncthreads();

    const float* sac = sa[cur];
    const float* sbc = sb[cur];
#pragma unroll
    for (int k8 = 0; k8 < 8; ++k8) {
      const int k = 4 * k8;
      v2f a0 = *(const v2f*)&sac[(wr * 32 + lm) * A_STRIDE + k + 2 * hf];
      v2f a1 = *(const v2f*)&sac[(wr * 32 + 16 + lm) * A_STRIDE + k + 2 * hf];
      v2f b0, b1;
      b0.x = sbc[(k + 2 * hf) * B_STRIDE + wc * 32 + lm];
      b0.y = sbc[(k + 2 * hf + 1) * B_STRIDE + wc * 32 + lm];
      b1.x = sbc[(k + 2 * hf) * B_STRIDE + wc * 32 + 16 + lm];
      b1.y = sbc[(k + 2 * hf + 1) * B_STRIDE + wc * 32 + 16 + lm];
      acc00 = WMMA_F32X4(a0, b0, acc00);
      acc01 = WMMA_F32X4(a0, b1, acc01);
      acc10 = WMMA_F32X4(a1, b0, acc10);
      acc11 = WMMA_F32X4(a1, b1, acc11);
    }
    __syncthreads();  // everyone done reading cur before it is overwritten
    cur ^= 1;
  }

  // epilogue
  v8f* accs[2][2] = {{&acc00, &acc01}, {&acc10, &acc11}};
#pragma unroll
  for (int rt = 0; rt < 2; ++rt) {
#pragma unroll
    for (int ct = 0; ct < 2; ++ct) {
      const int col = colB + wc * 32 + ct * 16 + lm;
      const float bv = bias ? bias[col] : 0.f;
      v8f av = *accs[rt][ct];
#pragma unroll
      for (int r = 0; r < 8; ++r) {
        const int row = row0 + wr * 32 + rt * 16 + r + 8 * hf;
        float v = av[r] + bv;
        if (do_relu) v = fmaxf(v, 0.f);
        C[(size_t)row * Nd + col] = v;
      }
    }
  }
}

// ---------------------------------------------------------------------------
// Causal local attention, flash-style. One wave per (b, h, 16-query tile).
// q,k,v laid out [b, s, h, dh] (row-major [B*S, D]). Output same layout.
// ---------------------------------------------------------------------------
__global__ __launch_bounds__(32) void attn_local_kernel(
    const float* __restrict__ q, const float* __restrict__ k,
    const float* __restrict__ v, float* __restrict__ out)
{
  __shared__ float lds_p[16 * 17];
  const int lane = threadIdx.x & 31, lm = lane & 15, hf = lane >> 4;
  int blk = blockIdx.x;
  const int qt = blk % (Sq / 16); blk /= (Sq / 16);
  const int h = blk % Hh;
  const int b = blk / Hh;
  const int q0 = qt * 16;
  const float scale = 0.125f; // 1/sqrt(64)

  float m_run[8], l_run[8];
  v8f acc[4] = {};
#pragma unroll
  for (int r = 0; r < 8; ++r) { m_run[r] = -INFINITY; l_run[r] = 0.f; }

  const size_t qrow = ((size_t)(b * Sq + q0 + lm) * Hh + h) * DHd;

  for (int j = 0; j <= qt; ++j) {
    // S = (Q @ K^T) * scale, 16x16 tile via 16 fp32 WMMAs over dh=64
    v8f s = {};
#pragma unroll
    for (int kk = 0; kk < DHd; kk += 4) {
      v2f a, bf;
      a.x = q[qrow + kk + 2 * hf + 0];
      a.y = q[qrow + kk + 2 * hf + 1];
      size_t krow = ((size_t)(b * Sq + j * 16 + lm) * Hh + h) * DHd + kk + 2 * hf;
      bf.x = k[krow + 0];
      bf.y = k[krow + 1];
      s = WMMA_F32X4(a, bf, s);
    }
#pragma unroll
    for (int r = 0; r < 8; ++r) {
      float sv = s[r] * scale;
      sv = (j == qt && lm > r + 8 * hf) ? -1e9f : sv; // causal mask
      s[r] = sv;
    }
    // online softmax update per row (row = r + 8*hf, cols across 16-lane group)
#pragma unroll
    for (int r = 0; r < 8; ++r) {
      float rm = s[r];
#pragma unroll
      for (int off = 1; off < 16; off <<= 1)
        rm = fmaxf(rm, __shfl_xor(rm, off, 32));
      float mn = fmaxf(m_run[r], rm);
      float corr = __expf(m_run[r] - mn);
      float p = __expf(s[r] - mn);
      float rs = p;
#pragma unroll
      for (int off = 1; off < 16; off <<= 1)
        rs += __shfl_xor(rs, off, 32);
      l_run[r] = l_run[r] * corr + rs;
      m_run[r] = mn;
      lds_p[(r + 8 * hf) * 17 + lm] = p;
#pragma unroll
      for (int t = 0; t < 4; ++t) acc[t][r] *= corr;
    }
    __syncthreads();
    // acc += P @ V  (P re-read from LDS as A-fragment)
#pragma unroll
    for (int kk = 0; kk < 16; kk += 4) {
      v2f a;
      a.x = lds_p[lm * 17 + kk + 2 * hf + 0];
      a.y = lds_p[lm * 17 + kk + 2 * hf + 1];
      size_t vrow = ((size_t)(b * Sq + j * 16 + kk + 2 * hf) * Hh + h) * DHd + lm;
#pragma unroll
      for (int t = 0; t < 4; ++t) {
        v2f bf;
        bf.x = v[vrow + t * 16];
        bf.y = v[vrow + (size_t)Hh * DHd + t * 16];
        acc[t] = WMMA_F32X4(a, bf, acc[t]);
      }
    }
    __syncthreads();
  }

#pragma unroll
  for (int r = 0; r < 8; ++r) {
    float inv = 1.f / l_run[r];
    size_t orow = ((size_t)(b * Sq + q0 + r + 8 * hf) * Hh + h) * DHd;
#pragma unroll
    for (int t = 0; t < 4; ++t)
      out[orow + t * 16 + lm] = acc[t][r] * inv;
  }
}

// ---------------------------------------------------------------------------
// KNN memory attention. One wave per query (b,h,s); 4 waves / block.
// Each lane owns 128 of M=4096 slots; 32 rounds of wave argmax (top-k),
// then softmax over top-32 and weighted gather of mem_v.
// ---------------------------------------------------------------------------
__global__ __launch_bounds__(128) void attn_mem_kernel(
    const float* __restrict__ q, const float* __restrict__ mem_k,
    const float* __restrict__ mem_v, float* __restrict__ out)
{
  __shared__ float qsm[4][64];
  __shared__ float tvs[4][32];
  __shared__ int   tis[4][32];
  const int w = threadIdx.x >> 5, lane = threadIdx.x & 31;
  int gq = blockIdx.x * 4 + w;
  const int s = gq % Sq; gq /= Sq;
  const int h = gq % Hh;
  const int b = gq / Hh;
  const float scale = 0.125f;

  const size_t qoff = ((size_t)(b * Sq + s) * Hh + h) * DHd;
  qsm[w][lane]      = q[qoff + lane];
  qsm[w][lane + 32] = q[qoff + lane + 32];
  __syncthreads();

  float sims[128];
  for (int i = 0; i < 128; ++i) {
    const int m = lane + 32 * i;
    const float* kr = mem_k + ((size_t)(b * Mm + m) * Hh + h) * DHd;
    float d = 0.f;
#pragma unroll 8
    for (int c = 0; c < 64; ++c) d += qsm[w][c] * kr[c];
    sims[i] = d * scale;
  }

  for (int r = 0; r < Kt; ++r) {
    float best = -INFINITY; int bidx = 0x7fffffff;
    for (int i = 0; i < 128; ++i) {
      float vv = sims[i];
      if (vv > best) { best = vv; bidx = lane + 32 * i; } // first hit = lowest idx
    }
    for (int off = 16; off; off >>= 1) {
      float ov = __shfl_xor(best, off, 32);
      int   oi = __shfl_xor(bidx, off, 32);
      if (ov > best || (ov == best && oi < bidx)) { best = ov; bidx = oi; }
    }
    if (lane == 0) { tvs[w][r] = best; tis[w][r] = bidx; }
    if (lane == (bidx & 31)) sims[bidx >> 5] = -INFINITY;
  }
  __syncthreads();

  float maxv = tvs[w][0];
  float sum = 0.f, a0 = 0.f, a1 = 0.f;
  for (int r = 0; r < Kt; ++r) {
    float wgt = __expf(tvs[w][r] - maxv);
    sum += wgt;
    const float* vr = mem_v + ((size_t)(b * Mm + tis[w][r]) * Hh + h) * DHd;
    a0 += wgt * vr[lane];
    a1 += wgt * vr[lane + 32];
  }
  float inv = 1.f / sum;
  out[qoff + lane]      = a0 * inv;
  out[qoff + lane + 32] = a1 * inv;
}

// ---------------------------------------------------------------------------
// Per-head sigmoid gate: out = g*o_mem + (1-g)*o_local   (layout [b,s,h,dh])
// ---------------------------------------------------------------------------
__global__ void combine_gate_kernel(const float* __restrict__ om,
                                    const float* __restrict__ ol,
                                    const float* __restrict__ gate,
                                    float* __restrict__ out, int n)
{
  int i = blockIdx.x * 256 + threadIdx.x;
  if (i >= n) return;
  int h = (i >> 6) & (Hh - 1);
  float g = 1.f / (1.f + __expf(-gate[h]));
  out[i] = g * om[i] + (1.f - g) * ol[i];
}

// ---------------------------------------------------------------------------
// out = LayerNorm(a + bres) * g + bb. One wave per row of D=1024.
// ---------------------------------------------------------------------------
__global__ __launch_bounds__(128) void add_ln_kernel(
    const float* __restrict__ a, const float* __restrict__ bres,
    const float* __restrict__ g, const float* __restrict__ bb,
    float* __restrict__ out)
{
  const int w = threadIdx.x >> 5, lane = threadIdx.x & 31;
  const int row = blockIdx.x * 4 + w;
  const size_t base = (size_t)row * Dm;
  float vals[32];
  float sum = 0.f, sq = 0.f;
#pragma unroll
  for (int i = 0; i < 32; ++i) {
    float x = a[base + lane + 32 * i] + bres[base + lane + 32 * i];
    vals[i] = x; sum += x; sq += x * x;
  }
#pragma unroll
  for (int off = 1; off < 32; off <<= 1) {
    sum += __shfl_xor(sum, off, 32);
    sq  += __shfl_xor(sq,  off, 32);
  }
  float mu = sum * (1.f / Dm);
  float var = sq * (1.f / Dm) - mu * mu;
  float rstd = rsqrtf(var + 1e-5f);
#pragma unroll
  for (int i = 0; i < 32; ++i) {
    int d = lane + 32 * i;
    out[base + d] = (vals[i] - mu) * rstd * g[d] + bb[d];
  }
}

// ---------------------------------------------------------------------------
extern "C" void kernel_launch(void* const* d_in, const int* in_sizes, int n_in,
                              void* d_out, int out_size, void* d_ws, size_t ws_size,
                              hipStream_t stream)
{
  const float* x      = (const float*)d_in[0];
  const float* wq     = (const float*)d_in[1];
  const float* wk     = (const float*)d_in[2];
  const float* wv     = (const float*)d_in[3];
  const float* wo     = (const float*)d_in[4];
  const float* gate   = (const float*)d_in[5];
  const float* mem_k  = (const float*)d_in[6];
  const float* mem_v  = (const float*)d_in[7];
  const float* ln2_g  = (const float*)d_in[8];
  const float* ln2_b  = (const float*)d_in[9];
  const float* ffn_w1 = (const float*)d_in[10];
  const float* ffn_b1 = (const float*)d_in[11];
  const float* ffn_w2 = (const float*)d_in[12];
  const float* ffn_b2 = (const float*)d_in[13];
  const float* ln3_g  = (const float*)d_in[14];
  const float* ln3_b  = (const float*)d_in[15];

  float* ws = (float*)d_ws;
  const size_t NT = (size_t)Bz * Sq * Dm;       // 4M floats per activation
  float* qb  = ws;                               // q        [B*S, D]
  float* kb  = ws + NT;                          // k / wo-out / reuse
  float* vb  = ws + 2 * NT;                      // v / x1   / reuse
  float* ol  = ws + 3 * NT;                      // o_local, then combined
  float* om  = ws + 4 * NT;                      // o_mem
  float* h1  = ws + 5 * NT;                      // ffn hidden [B*S, F] (16M floats)

  const int BSr = Bz * Sq;                       // 4096 rows

  // QKV projections
  gemm_wmma_f32<<<dim3(BSr / 64, Dm / 128), 256, 0, stream>>>(x, wq, nullptr, qb, BSr, Dm, Dm, 0);
  gemm_wmma_f32<<<dim3(BSr / 64, Dm / 128), 256, 0, stream>>>(x, wk, nullptr, kb, BSr, Dm, Dm, 0);
  gemm_wmma_f32<<<dim3(BSr / 64, Dm / 128), 256, 0, stream>>>(x, wv, nullptr, vb, BSr, Dm, Dm, 0);

  // Local causal attention -> ol
  attn_local_kernel<<<Bz * Hh * (Sq / 16), 32, 0, stream>>>(qb, kb, vb, ol);

  // KNN memory attention -> om
  attn_mem_kernel<<<(Bz * Hh * Sq) / 4, 128, 0, stream>>>(qb, mem_k, mem_v, om);

  // gate combine (in place into ol)
  combine_gate_kernel<<<(int)(NT / 256), 256, 0, stream>>>(om, ol, gate, ol, (int)NT);

  // attn_out = combined @ wo -> kb
  gemm_wmma_f32<<<dim3(BSr / 64, Dm / 128), 256, 0, stream>>>(ol, wo, nullptr, kb, BSr, Dm, Dm, 0);

  // x1 = LN(attn_out + x) -> vb
  add_ln_kernel<<<BSr / 4, 128, 0, stream>>>(kb, x, ln2_g, ln2_b, vb);

  // h1 = relu(x1 @ w1 + b1)
  gemm_wmma_f32<<<dim3(BSr / 64, Ff / 128), 256, 0, stream>>>(vb, ffn_w1, ffn_b1, h1, BSr, Ff, Dm, 1);

  // f = h1 @ w2 + b2 -> qb
  gemm_wmma_f32<<<dim3(BSr / 64, Dm / 128), 256, 0, stream>>>(h1, ffn_w2, ffn_b2, qb, BSr, Dm, Ff, 0);

  // out = LN(f + x1)
  add_ln_kernel<<<BSr / 4, 128, 0, stream>>>(qb, vb, ln3_g, ln3_b, (float*)d_out);
}